// TransformerModel_76510547411535
// MI455X (gfx1250) — compile-verified
//
#include <hip/hip_runtime.h>
#include <hip/hip_bf16.h>

#define B_    8
#define S_    1025
#define DIN_  512
#define H_    512
#define NH_   8
#define HD_   64
#define L_    4
#define FFN_  1024
#define HOPS_ 4
#define SEQ_  256
#define C_    40
#define MN_   (B_*S_)          /* 8200 token rows */
#define SCALE_ 0.125f          /* 64^-0.5 */
#define EPS_   1e-5f
#define QT_   ((S_+15)/16)     /* 65 tiles of 16 */

typedef float v2f __attribute__((ext_vector_type(2)));
typedef float v8f __attribute__((ext_vector_type(8)));

static __device__ __forceinline__ v8f wmma4(v2f a, v2f b, v8f c) {
  // D = A(16x4,f32) * B(4x16,f32) + C(16x16,f32)
  return __builtin_amdgcn_wmma_f32_16x16x4_f32(false, a, false, b, (short)0, c, false, false);
}

static __device__ __forceinline__ float gelu_exact(float x) {
  return 0.5f * x * (1.0f + erff(x * 0.70710678118654752f));
}

// ---------------------------------------------------------------------------
// Generic GEMM: C[M,N] = act(A[M,K] @ W[K,N] + bias[N]).  act: 0=none 1=gelu
// One wave computes a 64x64 tile via 4x4 grid of 16x16 fp32 WMMA accumulators.
// Rows of A are clamped (M need not be a multiple of 16); stores are masked.
// ---------------------------------------------------------------------------
__global__ __launch_bounds__(32) void gemm_wmma_kernel(
    const float* __restrict__ A, const float* __restrict__ W,
    const float* __restrict__ bias, float* __restrict__ Cmat,
    int M, int K, int N, int act)
{
  const int lane = threadIdx.x;
  const int half = lane >> 4;
  const int l16  = lane & 15;
  const int m0 = blockIdx.x * 64;
  const int n0 = blockIdx.y * 64;

  v8f zero = {0.f,0.f,0.f,0.f,0.f,0.f,0.f,0.f};
  v8f acc[4][4];
#pragma unroll
  for (int mt = 0; mt < 4; ++mt)
#pragma unroll
    for (int nt = 0; nt < 4; ++nt) acc[mt][nt] = zero;

  size_t arow[4];
#pragma unroll
  for (int mt = 0; mt < 4; ++mt) {
    int r = m0 + mt*16 + l16;
    if (r > M - 1) r = M - 1;
    arow[mt] = (size_t)r * (size_t)K;
  }

  for (int k = 0; k < K; k += 4) {
    v2f a[4], b[4];
#pragma unroll
    for (int mt = 0; mt < 4; ++mt)
      a[mt] = *(const v2f*)(A + arow[mt] + k + 2*half);   // A layout: vgpr0=K(2*half), vgpr1=+1
#pragma unroll
    for (int nt = 0; nt < 4; ++nt) {
      const float* wp = W + (size_t)(k + 2*half) * N + n0 + nt*16 + l16;
      b[nt].x = wp[0];                                    // B layout mirrors A along K
      b[nt].y = wp[N];
    }
#pragma unroll
    for (int mt = 0; mt < 4; ++mt)
#pragma unroll
      for (int nt = 0; nt < 4; ++nt)
        acc[mt][nt] = wmma4(a[mt], b[nt], acc[mt][nt]);
  }

#pragma unroll
  for (int nt = 0; nt < 4; ++nt) {
    const int col = n0 + nt*16 + l16;
    const float bv = bias ? bias[col] : 0.f;
#pragma unroll
    for (int mt = 0; mt < 4; ++mt) {
#pragma unroll
      for (int v = 0; v < 8; ++v) {
        int row = m0 + mt*16 + v + 8*half;                // C layout: row = vgpr + 8*(lane/16)
        if (row < M) {
          float x = acc[mt][nt][v] + bv;
          if (act == 1) x = gelu_exact(x);
          Cmat[(size_t)row * N + col] = x;
        }
      }
    }
  }
}

// ---------------------------------------------------------------------------
// Flash attention: scores = SCALE*(q.kT + qp.kpT), online softmax, O = P @ V.
// One wave handles one (b, head, 16-row q tile).  All matmuls are fp32 WMMA.
// ---------------------------------------------------------------------------
__global__ __launch_bounds__(32) void attn_kernel(
    const float* __restrict__ Q,  const float* __restrict__ Km,
    const float* __restrict__ V,  const float* __restrict__ QP,
    const float* __restrict__ KP, float* __restrict__ O)
{
  __shared__ float pls[16 * 17];
  const int lane = threadIdx.x;
  const int half = lane >> 4;
  const int l16  = lane & 15;
  const int q0 = blockIdx.x * 16;
  const int h  = blockIdx.y;
  const int b  = blockIdx.z;

  int qr = q0 + l16; if (qr > S_ - 1) qr = S_ - 1;
  const size_t qbase = ((size_t)(b * S_ + qr)) * H_ + h * HD_;

  v2f qa[16], qpa[16];
#pragma unroll
  for (int j = 0; j < 16; ++j) {
    qa[j]  = *(const v2f*)(Q  + qbase + 4*j + 2*half);
    qpa[j] = *(const v2f*)(QP + qbase + 4*j + 2*half);
  }

  v8f zero = {0.f,0.f,0.f,0.f,0.f,0.f,0.f,0.f};
  v8f oacc[4] = {zero, zero, zero, zero};
  float m[8], lsum[8];
#pragma unroll
  for (int v = 0; v < 8; ++v) { m[v] = -__builtin_inff(); lsum[v] = 0.f; }

  for (int kt = 0; kt < QT_; ++kt) {
    int kr = kt*16 + l16; if (kr > S_ - 1) kr = S_ - 1;
    const size_t kbase = ((size_t)(b * S_ + kr)) * H_ + h * HD_;

    v8f sc = zero;
#pragma unroll
    for (int j = 0; j < 16; ++j) {
      v2f kb = *(const v2f*)(Km + kbase + 4*j + 2*half);
      sc = wmma4(qa[j], kb, sc);
    }
#pragma unroll
    for (int j = 0; j < 16; ++j) {
      v2f kpb = *(const v2f*)(KP + kbase + 4*j + 2*half);
      sc = wmma4(qpa[j], kpb, sc);
    }

    const bool valid = (kt*16 + l16) < S_;   // this lane's kv column
#pragma unroll
    for (int v = 0; v < 8; ++v) {
      float s = sc[v] * SCALE_;
      sc[v] = valid ? s : -1e30f;
    }

    // online softmax (row = v + 8*half, reductions within each 16-lane half)
#pragma unroll
    for (int v = 0; v < 8; ++v) {
      float r = sc[v];
      r = fmaxf(r, __shfl_xor(r, 1, 32));
      r = fmaxf(r, __shfl_xor(r, 2, 32));
      r = fmaxf(r, __shfl_xor(r, 4, 32));
      r = fmaxf(r, __shfl_xor(r, 8, 32));
      float mn = fmaxf(m[v], r);
      float al = expf(m[v] - mn);
      float pv = expf(sc[v] - mn);
      float rs = pv;
      rs += __shfl_xor(rs, 1, 32);
      rs += __shfl_xor(rs, 2, 32);
      rs += __shfl_xor(rs, 4, 32);
      rs += __shfl_xor(rs, 8, 32);
      lsum[v] = lsum[v] * al + rs;
      m[v] = mn;
#pragma unroll
      for (int nt = 0; nt < 4; ++nt) oacc[nt][v] *= al;
      pls[(v + 8*half) * 17 + l16] = pv;     // P in row-major LDS (C layout -> matrix)
    }
    __syncthreads();

    // O += P(16x16) @ V(16x64): P re-read in A layout from LDS
#pragma unroll
    for (int j = 0; j < 4; ++j) {
      v2f pa;
      pa.x = pls[l16 * 17 + 4*j + 2*half];
      pa.y = pls[l16 * 17 + 4*j + 2*half + 1];
      int vr0 = kt*16 + 4*j + 2*half;     if (vr0 > S_ - 1) vr0 = S_ - 1;
      int vr1 = kt*16 + 4*j + 2*half + 1; if (vr1 > S_ - 1) vr1 = S_ - 1;
      const size_t vb0 = ((size_t)(b * S_ + vr0)) * H_ + h * HD_;
      const size_t vb1 = ((size_t)(b * S_ + vr1)) * H_ + h * HD_;
#pragma unroll
      for (int nt = 0; nt < 4; ++nt) {
        v2f vb;
        vb.x = V[vb0 + nt*16 + l16];
        vb.y = V[vb1 + nt*16 + l16];
        oacc[nt] = wmma4(pa, vb, oacc[nt]);
      }
    }
    __syncthreads();
  }

#pragma unroll
  for (int v = 0; v < 8; ++v) {
    float inv = 1.0f / lsum[v];
    int row = q0 + v + 8*half;
    if (row < S_) {
#pragma unroll
      for (int nt = 0; nt < 4; ++nt)
        O[((size_t)(b * S_ + row)) * H_ + h * HD_ + nt*16 + l16] = oacc[nt][v] * inv;
    }
  }
}

// ---------------------------------------------------------------------------
// out[row] = LN(A[row] + Badd[row]) * g + beta   (one wave per row, H=512)
// ---------------------------------------------------------------------------
__global__ __launch_bounds__(32) void ln_add_kernel(
    const float* __restrict__ A, const float* __restrict__ Badd,
    const float* __restrict__ g, const float* __restrict__ beta,
    float* __restrict__ out)
{
  const int row = blockIdx.x;
  const int lane = threadIdx.x;
  float x[16];
  float s = 0.f;
#pragma unroll
  for (int i = 0; i < 16; ++i) {
    int c = lane + i*32;
    x[i] = A[(size_t)row*H_ + c] + Badd[(size_t)row*H_ + c];
    s += x[i];
  }
#pragma unroll
  for (int off = 1; off < 32; off <<= 1) s += __shfl_xor(s, off, 32);
  const float mean = s * (1.0f / H_);
  float vv = 0.f;
#pragma unroll
  for (int i = 0; i < 16; ++i) { float d = x[i] - mean; vv += d*d; }
#pragma unroll
  for (int off = 1; off < 32; off <<= 1) vv += __shfl_xor(vv, off, 32);
  const float rs = rsqrtf(vv * (1.0f / H_) + EPS_);
#pragma unroll
  for (int i = 0; i < 16; ++i) {
    int c = lane + i*32;
    out[(size_t)row*H_ + c] = (x[i] - mean) * rs * g[c] + beta[c];
  }
}

// --------------------------- graph-attention head ---------------------------
__global__ __launch_bounds__(256) void node_neigh_kernel(
    const float* __restrict__ X, float* __restrict__ node, float* __restrict__ neigh)
{
  const int bh = blockIdx.x, b = bh / HOPS_, hop = bh % HOPS_;
  for (int c = threadIdx.x; c < H_; c += 256) {
    const float* base = X + ((size_t)(b*S_ + 1 + hop*SEQ_)) * H_ + c;
    float s = 0.f;
    for (int t = 0; t < SEQ_; ++t) s += base[(size_t)t * H_];
    neigh[(size_t)bh*H_ + c] = s;
    if (hop == 0) node[(size_t)b*H_ + c] = X[((size_t)b*S_)*H_ + c];
  }
}

__global__ __launch_bounds__(256) void att_logit_kernel(
    const float* __restrict__ node, const float* __restrict__ neigh,
    const float* __restrict__ attn_w, const float* __restrict__ attn_b,
    float* __restrict__ logits)
{
  __shared__ float red[256];
  const int bh = blockIdx.x, b = bh / HOPS_, tid = threadIdx.x;
  float s = 0.f;
  for (int c = tid; c < H_; c += 256)
    s += node[(size_t)b*H_ + c] * attn_w[c] + neigh[(size_t)bh*H_ + c] * attn_w[H_ + c];
  red[tid] = s; __syncthreads();
  for (int off = 128; off > 0; off >>= 1) { if (tid < off) red[tid] += red[tid+off]; __syncthreads(); }
  if (tid == 0) logits[bh] = red[0] + attn_b[0];
}

__global__ __launch_bounds__(256) void head_final_kernel(
    const float* __restrict__ node, const float* __restrict__ neigh,
    const float* __restrict__ logits, const float* __restrict__ fn_g,
    const float* __restrict__ fn_b, const float* __restrict__ cls_w,
    const float* __restrict__ cls_b, float* __restrict__ out)
{
  __shared__ float w4[HOPS_];
  __shared__ float aggs[H_];
  __shared__ float red[256];
  __shared__ float z[C_];
  __shared__ float stat;
  const int b = blockIdx.x, tid = threadIdx.x;

  if (tid == 0) {
    float mx = -__builtin_inff();
    for (int h = 0; h < HOPS_; ++h) mx = fmaxf(mx, logits[b*HOPS_ + h]);
    float ss = 0.f;
    for (int h = 0; h < HOPS_; ++h) { float e = expf(logits[b*HOPS_ + h] - mx); w4[h] = e; ss += e; }
    for (int h = 0; h < HOPS_; ++h) w4[h] /= ss;
  }
  __syncthreads();
  for (int c = tid; c < H_; c += 256) {
    float a = 0.f;
    for (int h = 0; h < HOPS_; ++h) a += neigh[(size_t)(b*HOPS_ + h)*H_ + c] * w4[h];
    aggs[c] = a;
  }
  __syncthreads();
  red[tid] = aggs[tid] + aggs[tid + 256]; __syncthreads();
  for (int off = 128; off > 0; off >>= 1) { if (tid < off) red[tid] += red[tid+off]; __syncthreads(); }
  const float mean = red[0] / H_;
  __syncthreads();
  { float d0 = aggs[tid] - mean, d1 = aggs[tid + 256] - mean; red[tid] = d0*d0 + d1*d1; }
  __syncthreads();
  for (int off = 128; off > 0; off >>= 1) { if (tid < off) red[tid] += red[tid+off]; __syncthreads(); }
  const float rsv = rsqrtf(red[0] / H_ + EPS_);
  __syncthreads();
  for (int c = tid; c < H_; c += 256) {
    float an = (aggs[c] - mean) * rsv * fn_g[c] + fn_b[c];
    float o  = node[(size_t)b*H_ + c] + an;
    aggs[c] = (o > 0.f) ? o : 0.01f * o;          // leaky_relu
  }
  __syncthreads();
  for (int j = tid; j < C_; j += 256) {
    float a = cls_b[j];
    for (int c = 0; c < H_; ++c) a += aggs[c] * cls_w[(size_t)c*C_ + j];
    z[j] = a;
  }
  __syncthreads();
  if (tid == 0) {
    float mx = -__builtin_inff();
    for (int j = 0; j < C_; ++j) mx = fmaxf(mx, z[j]);
    float ss = 0.f;
    for (int j = 0; j < C_; ++j) ss += expf(z[j] - mx);
    stat = mx + logf(ss);
  }
  __syncthreads();
  for (int j = tid; j < C_; j += 256) out[(size_t)b*C_ + j] = z[j] - stat;
}

// ---------------------------------------------------------------------------
static void launch_gemm(const float* A, const float* W, const float* bias, float* Cm,
                        int M, int K, int N, int act, hipStream_t s)
{
  dim3 grid((M + 63) / 64, N / 64);
  gemm_wmma_kernel<<<grid, 32, 0, s>>>(A, W, bias, Cm, M, K, N, act);
}

extern "C" void kernel_launch(void* const* d_in, const int* in_sizes, int n_in,
                              void* d_out, int out_size, void* d_ws, size_t ws_size,
                              hipStream_t stream)
{
  const float* data   = (const float*)d_in[0];
  const float* pe     = (const float*)d_in[1];
  const float* emb_w  = (const float*)d_in[2];
  const float* emb_b  = (const float*)d_in[3];
  const float* q_w    = (const float*)d_in[4];
  const float* q_b    = (const float*)d_in[5];
  const float* k_w    = (const float*)d_in[6];
  const float* k_b    = (const float*)d_in[7];
  const float* v_w    = (const float*)d_in[8];
  const float* v_b    = (const float*)d_in[9];
  const float* qpe_w  = (const float*)d_in[10];
  const float* qpe_b  = (const float*)d_in[11];
  const float* kpe_w  = (const float*)d_in[12];
  const float* kpe_b  = (const float*)d_in[13];
  const float* out_w  = (const float*)d_in[14];
  const float* out_b  = (const float*)d_in[15];
  const float* ln1_g  = (const float*)d_in[16];
  const float* ln1_b  = (const float*)d_in[17];
  const float* f1_w   = (const float*)d_in[18];
  const float* f1_b   = (const float*)d_in[19];
  const float* f2_w   = (const float*)d_in[20];
  const float* f2_b   = (const float*)d_in[21];
  const float* ln2_g  = (const float*)d_in[22];
  const float* ln2_b  = (const float*)d_in[23];
  const float* attn_w = (const float*)d_in[24];
  const float* attn_b = (const float*)d_in[25];
  const float* fn_g   = (const float*)d_in[26];
  const float* fn_b   = (const float*)d_in[27];
  const float* cls_w  = (const float*)d_in[28];
  const float* cls_b  = (const float*)d_in[29];

  float* ws = (float*)d_ws;
  const size_t MH = (size_t)MN_ * H_;
  float* X   = ws;
  float* Qb  = ws + 1*MH;
  float* Kb  = ws + 2*MH;
  float* Vb  = ws + 3*MH;
  float* QPb = ws + 4*MH;
  float* KPb = ws + 5*MH;
  float* Ob  = ws + 6*MH;
  float* Yb  = ws + 7*MH;
  float* F1  = Qb;              // MN_ x FFN aliases Q+K slots (dead during FFN)
  float* T1  = QPb;             // out-proj result aliases QP slot
  float* F2  = Vb;              // FFN output aliases V slot
  float* node   = ws + 8*MH;
  float* neigh  = node  + (size_t)B_ * H_;
  float* logits = neigh + (size_t)B_ * HOPS_ * H_;

  // x = data @ emb_w + emb_b
  launch_gemm(data, emb_w, emb_b, X, MN_, DIN_, H_, 0, stream);

  for (int i = 0; i < L_; ++i) {
    const size_t wHH = (size_t)i * H_ * H_;
    launch_gemm(X,  q_w   + wHH, q_b   + i*H_, Qb,  MN_, H_, H_, 0, stream);
    launch_gemm(X,  k_w   + wHH, k_b   + i*H_, Kb,  MN_, H_, H_, 0, stream);
    launch_gemm(X,  v_w   + wHH, v_b   + i*H_, Vb,  MN_, H_, H_, 0, stream);
    launch_gemm(pe, qpe_w + wHH, qpe_b + i*H_, QPb, MN_, H_, H_, 0, stream);
    launch_gemm(pe, kpe_w + wHH, kpe_b + i*H_, KPb, MN_, H_, H_, 0, stream);

    attn_kernel<<<dim3(QT_, NH_, B_), 32, 0, stream>>>(Qb, Kb, Vb, QPb, KPb, Ob);

    launch_gemm(Ob, out_w + wHH, out_b + i*H_, T1, MN_, H_, H_, 1, stream);       // gelu fused
    ln_add_kernel<<<MN_, 32, 0, stream>>>(X, T1, ln1_g + i*H_, ln1_b + i*H_, Yb);

    launch_gemm(Yb, f1_w + (size_t)i*H_*FFN_, f1_b + i*FFN_, F1, MN_, H_, FFN_, 1, stream);
    launch_gemm(F1, f2_w + (size_t)i*FFN_*H_, f2_b + i*H_,   F2, MN_, FFN_, H_, 0, stream);
    ln_add_kernel<<<MN_, 32, 0, stream>>>(Yb, F2, ln2_g + i*H_, ln2_b + i*H_, X);
  }

  node_neigh_kernel<<<B_*HOPS_, 256, 0, stream>>>(X, node, neigh);
  att_logit_kernel<<<B_*HOPS_, 256, 0, stream>>>(node, neigh, attn_w, attn_b, logits);
  head_final_kernel<<<B_, 256, 0, stream>>>(node, neigh, logits, fn_g, fn_b,
                                            cls_w, cls_b, (float*)d_out);
}